// GCN_49950469653067
// MI455X (gfx1250) — compile-verified
//
#include <hip/hip_runtime.h>

#define NNODES   50000
#define NEDGES   800000
#define NGRAPHS  512
#define INDIM    128
#define HID      64
#define NCLS     2

typedef __attribute__((ext_vector_type(2))) float v2f;
typedef __attribute__((ext_vector_type(8))) float v8f;

// ---------------- degree / normalization ----------------
__global__ void deg_init_kernel(float* deg, int n) {
    int i = blockIdx.x * blockDim.x + threadIdx.x;
    if (i < n) deg[i] = 1.0f;  // self-loop contribution
}

__global__ void deg_count_kernel(const int* __restrict__ dst, float* deg, int ne) {
    int i = blockIdx.x * blockDim.x + threadIdx.x;
    if (i < ne) atomicAdd(&deg[dst[i]], 1.0f);
}

__global__ void deg_rsqrt_kernel(float* deg, int n) {
    int i = blockIdx.x * blockDim.x + threadIdx.x;
    if (i < n) deg[i] = rsqrtf(deg[i]);  // in-place -> dinv (deg >= 1 always)
}

// ---------------- WMMA fp32 GEMM: out[nrows x 64] = (relu?)A[nrows x K] @ W[K x 64] ----------------
// Block = 256 threads = 8 waves; wave w computes rows [tile*16, tile*16+16) x 64 cols using
// V_WMMA_F32_16X16X4_F32 (exact fp32). W is staged in LDS *k-pair interleaved*:
//   ldsW[((k>>1)*64 + n)*2 + (k&1)] = W[k][n]
// so each B fragment {W[ka][n], W[ka+1][n]} is one even-aligned ds_load_b64 (no repack movs).
template<int K, bool RELU>
__global__ void gemm_wmma_kernel(const float* __restrict__ A, const float* __restrict__ W,
                                 float* __restrict__ out, int nrows) {
    __shared__ float ldsW[K * 64];
    for (int i = threadIdx.x; i < K * 64; i += blockDim.x) {
        int k = i >> 6, n = i & 63;
        ldsW[(((k >> 1) << 6) + n) * 2 + (k & 1)] = W[i];
    }
    __syncthreads();

    const int wave = threadIdx.x >> 5;
    const int lane = threadIdx.x & 31;
    const int half = lane >> 4;   // 0: lanes 0-15, 1: lanes 16-31
    const int lr   = lane & 15;
    const int tile = blockIdx.x * (blockDim.x >> 5) + wave;
    if (tile * 16 >= nrows) return;

    int row = tile * 16 + lr;
    if (row >= nrows) row = nrows - 1;          // clamp (keeps EXEC all-ones for WMMA)
    const long arow = (long)row * K;

    v8f acc0 = {}, acc1 = {}, acc2 = {}, acc3 = {};
    #pragma unroll
    for (int k0 = 0; k0 < K; k0 += 4) {
        const int ka = k0 + 2 * half;           // even
        float2 av = *(const float2*)&A[arow + ka];   // A[M=row][ka], A[M=row][ka+1]
        v2f a;
        a.x = RELU ? fmaxf(av.x, 0.f) : av.x;
        a.y = RELU ? fmaxf(av.y, 0.f) : av.y;
        const v2f* brow = (const v2f*)&ldsW[(ka >> 1) * 128];  // 64 interleaved k-pairs
        v2f b;
        b = brow[lr];
        acc0 = __builtin_amdgcn_wmma_f32_16x16x4_f32(false, a, false, b, (short)0, acc0, false, false);
        b = brow[lr + 16];
        acc1 = __builtin_amdgcn_wmma_f32_16x16x4_f32(false, a, false, b, (short)0, acc1, false, false);
        b = brow[lr + 32];
        acc2 = __builtin_amdgcn_wmma_f32_16x16x4_f32(false, a, false, b, (short)0, acc2, false, false);
        b = brow[lr + 48];
        acc3 = __builtin_amdgcn_wmma_f32_16x16x4_f32(false, a, false, b, (short)0, acc3, false, false);
    }

    v8f accs[4] = {acc0, acc1, acc2, acc3};
    #pragma unroll
    for (int t = 0; t < 4; ++t) {
        #pragma unroll
        for (int r = 0; r < 8; ++r) {
            int m = tile * 16 + r + 8 * half;  // C/D layout: VGPR r -> M=r+8*half, N=lr+16t
            if (m < nrows) out[(long)m * 64 + t * 16 + lr] = accs[t][r];
        }
    }
}

// ---------------- layer-3 GEMM (N=2, VALU): xw3[n][c] = sum_k relu(h2[n][k]) * W3[k][c] ----------------
__global__ void gemm_small_kernel(const float* __restrict__ h2, const float* __restrict__ W3,
                                  float* __restrict__ xw3, int n) {
    __shared__ float w3[HID * NCLS];
    if (threadIdx.x < HID * NCLS) w3[threadIdx.x] = W3[threadIdx.x];
    __syncthreads();
    int i = blockIdx.x * blockDim.x + threadIdx.x;
    if (i >= n) return;
    float s0 = 0.f, s1 = 0.f;
    const float* hrow = h2 + (long)i * HID;
    #pragma unroll
    for (int k = 0; k < HID; ++k) {
        float a = fmaxf(hrow[k], 0.f);
        s0 = fmaf(a, w3[k * 2 + 0], s0);
        s1 = fmaf(a, w3[k * 2 + 1], s1);
    }
    xw3[i * 2 + 0] = s0;
    xw3[i * 2 + 1] = s1;
}

// ---------------- aggregation: init with bias + self-loop, then edge scatter ----------------
__global__ void agg_init_kernel(const float* __restrict__ xw, const float* __restrict__ dinv,
                                const float* __restrict__ bias, float* __restrict__ h,
                                int n, int F, int logF) {
    int idx = blockIdx.x * blockDim.x + threadIdx.x;
    if (idx >= n * F) return;
    int node = idx >> logF;
    int f = idx & (F - 1);
    float di = dinv[node];
    h[idx] = bias[f] + xw[idx] * di * di;
}

// one wave per edge, 64 features (2 per lane): coalesced gather + float atomics at dst
__global__ void agg_edges64_kernel(const float* __restrict__ xw, const float* __restrict__ dinv,
                                   const int* __restrict__ src, const int* __restrict__ dst,
                                   float* __restrict__ h, int ne) {
    int wave = threadIdx.x >> 5;
    int lane = threadIdx.x & 31;
    long e = (long)blockIdx.x * (blockDim.x >> 5) + wave;
    if (e >= ne) return;
    int s = src[e], d = dst[e];
    float nrm = dinv[s] * dinv[d];
    float v0 = xw[(long)s * 64 + lane]      * nrm;
    float v1 = xw[(long)s * 64 + 32 + lane] * nrm;
    atomicAdd(&h[(long)d * 64 + lane],      v0);
    atomicAdd(&h[(long)d * 64 + 32 + lane], v1);
}

// thread per edge, 2 features (layer 3)
__global__ void agg_edges2_kernel(const float* __restrict__ xw, const float* __restrict__ dinv,
                                  const int* __restrict__ src, const int* __restrict__ dst,
                                  float* __restrict__ h, int ne) {
    int e = blockIdx.x * blockDim.x + threadIdx.x;
    if (e >= ne) return;
    int s = src[e], d = dst[e];
    float nrm = dinv[s] * dinv[d];
    atomicAdd(&h[d * 2 + 0], xw[s * 2 + 0] * nrm);
    atomicAdd(&h[d * 2 + 1], xw[s * 2 + 1] * nrm);
}

// ---------------- mean pool ----------------
__global__ void pool_init_kernel(float* ps, float* pc) {
    int i = blockIdx.x * blockDim.x + threadIdx.x;
    if (i < NGRAPHS * 2) ps[i] = 0.f;
    if (i < NGRAPHS) pc[i] = 0.f;
}

__global__ void pool_accum_kernel(const float* __restrict__ h3, const int* __restrict__ batch,
                                  float* ps, float* pc, int n) {
    int i = blockIdx.x * blockDim.x + threadIdx.x;
    if (i >= n) return;
    int b = batch[i];
    atomicAdd(&ps[b * 2 + 0], h3[i * 2 + 0]);
    atomicAdd(&ps[b * 2 + 1], h3[i * 2 + 1]);
    atomicAdd(&pc[b], 1.0f);
}

__global__ void pool_final_kernel(const float* __restrict__ ps, const float* __restrict__ pc,
                                  float* __restrict__ out) {
    int i = blockIdx.x * blockDim.x + threadIdx.x;
    if (i < NGRAPHS * 2) out[i] = ps[i] / fmaxf(pc[i >> 1], 1.0f);
}

extern "C" void kernel_launch(void* const* d_in, const int* in_sizes, int n_in,
                              void* d_out, int out_size, void* d_ws, size_t ws_size,
                              hipStream_t stream) {
    const float* x    = (const float*)d_in[0];
    const int*   eidx = (const int*)d_in[1];
    const int*   batch= (const int*)d_in[2];
    const float* W1   = (const float*)d_in[3];
    const float* b1   = (const float*)d_in[4];
    const float* W2   = (const float*)d_in[5];
    const float* b2   = (const float*)d_in[6];
    const float* W3   = (const float*)d_in[7];
    const float* b3   = (const float*)d_in[8];
    float* out = (float*)d_out;

    const int* src = eidx;            // edge_index[0]
    const int* dst = eidx + NEDGES;   // edge_index[1]

    float* ws   = (float*)d_ws;
    float* dinv = ws;                          // 50048
    float* bufA = ws + 50048;                  // 3,200,000 (xw1 -> xw2 -> xw3)
    float* bufB = bufA + 3200000;              // 3,200,000 (h1 -> h2 -> h3)
    float* ps   = bufB + 3200000;              // 1024
    float* pc   = ps + 1024;                   // 512

    const int T = 256;
    dim3 blk(T);

    // degrees -> dinv
    deg_init_kernel <<<(NNODES + T - 1) / T, blk, 0, stream>>>(dinv, NNODES);
    deg_count_kernel<<<(NEDGES + T - 1) / T, blk, 0, stream>>>(dst, dinv, NEDGES);
    deg_rsqrt_kernel<<<(NNODES + T - 1) / T, blk, 0, stream>>>(dinv, NNODES);

    const int mtiles = (NNODES + 15) / 16;           // 3125 (exact: 3125*16 == 50000)
    const int gblocks = (mtiles + 7) / 8;            // 391 (8 waves/block)
    const int eblocks64 = (NEDGES + 7) / 8;          // wave per edge

    // ---- layer 1: xw1 = x @ W1 ; h1 = b1 + self + edges ----
    gemm_wmma_kernel<INDIM, false><<<gblocks, blk, 0, stream>>>(x, W1, bufA, NNODES);
    agg_init_kernel <<<(NNODES * 64 + T - 1) / T, blk, 0, stream>>>(bufA, dinv, b1, bufB, NNODES, 64, 6);
    agg_edges64_kernel<<<eblocks64, blk, 0, stream>>>(bufA, dinv, src, dst, bufB, NEDGES);

    // ---- layer 2: xw2 = relu(h1) @ W2 ; h2 = b2 + self + edges ----
    gemm_wmma_kernel<HID, true><<<gblocks, blk, 0, stream>>>(bufB, W2, bufA, NNODES);
    agg_init_kernel <<<(NNODES * 64 + T - 1) / T, blk, 0, stream>>>(bufA, dinv, b2, bufB, NNODES, 64, 6);
    agg_edges64_kernel<<<eblocks64, blk, 0, stream>>>(bufA, dinv, src, dst, bufB, NEDGES);

    // ---- layer 3: xw3 = relu(h2) @ W3 (N=2) ; h3 = b3 + self + edges ----
    gemm_small_kernel<<<(NNODES + T - 1) / T, blk, 0, stream>>>(bufB, W3, bufA, NNODES);
    agg_init_kernel <<<(NNODES * 2 + T - 1) / T, blk, 0, stream>>>(bufA, dinv, b3, bufB, NNODES, 2, 1);
    agg_edges2_kernel<<<(NEDGES + T - 1) / T, blk, 0, stream>>>(bufA, dinv, src, dst, bufB, NEDGES);

    // ---- global mean pool ----
    pool_init_kernel <<<(NGRAPHS * 2 + T - 1) / T, blk, 0, stream>>>(ps, pc);
    pool_accum_kernel<<<(NNODES + T - 1) / T, blk, 0, stream>>>(bufB, batch, ps, pc, NNODES);
    pool_final_kernel<<<(NGRAPHS * 2 + T - 1) / T, blk, 0, stream>>>(ps, pc, out);
}